// Instance_Loss_16655883174016
// MI455X (gfx1250) — compile-verified
//
#include <hip/hip_runtime.h>
#include <stdint.h>

// Problem constants (match reference)
#define BB      8
#define NN      49104
#define MM      32
#define NUM_OBJ 90
#define NUM_ACT 48
#define ANNW    53          // 4 box + 1 label + 48 act one-hot
#define NXB     512         // grid.x blocks
#define TPB     512         // threads per block
#define WPB     (TPB / 32)  // 16 waves per block

typedef unsigned int v4u __attribute__((ext_vector_type(4)));
typedef int          v8i __attribute__((ext_vector_type(8)));
typedef int          v4i __attribute__((ext_vector_type(4)));

struct SmemT {
  float ann[MM * ANNW];   // MUST be first: TDM writes at this LDS offset
  float red[WPB][4];
};

__device__ __forceinline__ float wave_sum(float v) {
  #pragma unroll
  for (int off = 16; off; off >>= 1) v += __shfl_xor(v, off, 32);
  return v;
}

__global__ __launch_bounds__(TPB) void instance_loss_main(
    const float* __restrict__ act_cls,   // [B,N,48]
    const float* __restrict__ obj_cls,   // [B,N,90]
    const float* __restrict__ regr,      // [B,N,4]
    const float* __restrict__ anchors,   // [1,N,4] (y1,x1,y2,x2)
    const float* __restrict__ ann,       // [B,32,53] (x1,y1,x2,y2,label,act[48])
    float4* __restrict__ partials)       // [B, NXB] : {actSum, objSum, regSum, numPos}
{
  const int b    = blockIdx.y;
  const int lane = threadIdx.x & 31;
  const int wave = threadIdx.x >> 5;

  __shared__ SmemT sm;

  // ---- Stage this batch's 32x53 annotation block into LDS via the TDM ----
#if __has_builtin(__builtin_amdgcn_tensor_load_to_lds) && __has_builtin(__builtin_amdgcn_s_wait_tensorcnt)
  if (wave == 0) {
    uint64_t ga = (uint64_t)(uintptr_t)(ann + (size_t)b * (MM * ANNW));
    uint32_t ldsOff = (uint32_t)(uintptr_t)(&sm);      // addr[31:0] == LDS offset
    const uint32_t elems = MM * ANNW;                  // 1696 f32 = 6784 bytes
    v4u g0;
    g0.x = 1u;                                         // count=1, user descriptor
    g0.y = ldsOff;                                     // lds_addr (bytes)
    g0.z = (uint32_t)(ga & 0xFFFFFFFFu);               // global_addr[31:0]
    g0.w = (uint32_t)((ga >> 32) & 0x1FFFFFFu) | (2u << 30); // addr[56:32] | type=2
    v8i g1;
    g1[0] = (int)(2u << 16);                           // data_size = 2 (4 bytes)
    g1[1] = (int)(elems << 16);                        // tensor_dim0[15:0] in hi half
    g1[2] = 0;                                         // tensor_dim0 hi | tensor_dim1 lo (dim1 unused)
    g1[3] = (int)(elems << 16);                        // tile_dim0 = 1696 (1-D tile)
    g1[4] = 0;                                         // tile_dim1 = tile_dim2 = 0 (unused)
    g1[5] = (int)elems;                                // tensor_dim0_stride[31:0]
    g1[6] = 0;
    g1[7] = 0;
    v4i g2 = {0, 0, 0, 0};
    v4i g3 = {0, 0, 0, 0};
    v8i g4 = {0, 0, 0, 0, 0, 0, 0, 0};
    __builtin_amdgcn_tensor_load_to_lds(g0, g1, g2, g3, g4, 0);
    __builtin_amdgcn_s_wait_tensorcnt(0);
  }
#else
  for (int i = threadIdx.x; i < MM * ANNW; i += TPB)
    sm.ann[i] = ann[(size_t)b * (MM * ANNW) + i];
#endif
  __syncthreads();

  const size_t base_bn = (size_t)b * NN;
  float accO = 0.f, accA = 0.f, accR = 0.f, nposf = 0.f;

  const int waveGlobal = blockIdx.x * WPB + wave;
  const int stride     = NXB * WPB;

  for (int n = waveGlobal; n < NN; n += stride) {
    // Prefetch next iteration's classification rows (global_prefetch_b8)
    int n2 = n + stride;
    if (n2 < NN) {
      __builtin_prefetch(obj_cls + (base_bn + n2) * NUM_OBJ, 0, 0);
      __builtin_prefetch(act_cls + (base_bn + n2) * NUM_ACT, 0, 0);
    }

    // Anchor (y1,x1,y2,x2)
    const float ay1 = anchors[n * 4 + 0];
    const float ax1 = anchors[n * 4 + 1];
    const float ay2 = anchors[n * 4 + 2];
    const float ax2 = anchors[n * 4 + 3];
    const float aw = ax2 - ax1;
    const float ah = ay2 - ay1;

    // Lane m computes IoU with GT box m (M == 32 == wave32)
    const float* gb = &sm.ann[lane * ANNW];
    const float b0 = gb[0], b1 = gb[1], b2 = gb[2], b3 = gb[3];
    const float lbl = gb[4];
    float iou = -1.f;
    if (lbl != -1.f) {
      float iw = fminf(ax2, b2) - fmaxf(ax1, b0);
      float ih = fminf(ay2, b3) - fmaxf(ay1, b1);
      iw = fmaxf(iw, 0.f);
      ih = fmaxf(ih, 0.f);
      const float inter = iw * ih;
      const float areab = (b2 - b0) * (b3 - b1);
      const float ua = fmaxf(aw * ah + areab - inter, 1e-8f);
      iou = inter / ua;
    }

    // Wave argmax with first-index tiebreak (matches jnp.argmax)
    int arg = lane;
    #pragma unroll
    for (int off = 16; off; off >>= 1) {
      float oi = __shfl_xor(iou, off, 32);
      int   oa = __shfl_xor(arg, off, 32);
      if (oi > iou || (oi == iou && oa < arg)) { iou = oi; arg = oa; }
    }

    const bool pos = (iou >= 0.5f);
    const bool neg = (iou <  0.4f);

    if (pos || neg) {
      const int albl = pos ? (int)sm.ann[arg * ANNW + 4] : -1;

      // Focal obj: unified form loss = af*(1-x)^2*(-log x), x = t? p : 1-p
      const size_t obase = (base_bn + n) * NUM_OBJ;
      for (int c = lane; c < NUM_OBJ; c += 32) {
        float p = obj_cls[obase + c];
        p = fminf(fmaxf(p, 1e-4f), 1.f - 1e-4f);
        const bool t1 = pos && (c == albl);
        const float x   = t1 ? p : (1.f - p);
        const float af  = t1 ? 0.25f : 0.75f;
        const float omx = 1.f - x;
        accO += af * omx * omx * (-__logf(x));
      }

      // Focal act
      const size_t abase = (base_bn + n) * NUM_ACT;
      for (int c = lane; c < NUM_ACT; c += 32) {
        float p = act_cls[abase + c];
        p = fminf(fmaxf(p, 1e-4f), 1.f - 1e-4f);
        const float t  = pos ? sm.ann[arg * ANNW + 5 + c] : 0.f;
        const bool t1  = (t == 1.f);
        const float x   = t1 ? p : (1.f - p);
        const float af  = t1 ? 0.25f : 0.75f;
        const float omx = 1.f - x;
        accA += af * omx * omx * (-__logf(x));
      }
    }

    if (pos) {
      nposf += 1.f;  // wave-uniform; lane 0's copy is used
      if (lane < 4) {
        const float* ab = &sm.ann[arg * ANNW];   // assigned box x1,y1,x2,y2
        float gw = ab[2] - ab[0];
        float gh = ab[3] - ab[1];
        const float gcx = ab[0] + 0.5f * gw;     // centers from UNCLIPPED w/h
        const float gcy = ab[1] + 0.5f * gh;
        gw = fmaxf(gw, 1.f);
        gh = fmaxf(gh, 1.f);
        const float acx = ax1 + 0.5f * aw;
        const float acy = ay1 + 0.5f * ah;
        float tgt;
        if      (lane == 0) tgt = (gcy - acy) / ah;   // tdy
        else if (lane == 1) tgt = (gcx - acx) / aw;   // tdx
        else if (lane == 2) tgt = __logf(gh / ah);    // tdh
        else                tgt = __logf(gw / aw);    // tdw
        const float r = regr[(base_bn + n) * 4 + lane];
        const float d = fabsf(tgt - r);
        accR += (d <= (1.f / 9.f)) ? 4.5f * d * d : (d - 0.5f / 9.f);
      }
    }
  }

  // Per-wave reduce, then per-block reduce (fixed order -> deterministic)
  accO = wave_sum(accO);
  accA = wave_sum(accA);
  accR = wave_sum(accR);
  if (lane == 0) {
    sm.red[wave][0] = accA;
    sm.red[wave][1] = accO;
    sm.red[wave][2] = accR;
    sm.red[wave][3] = nposf;
  }
  __syncthreads();
  if (threadIdx.x == 0) {
    float4 o = {0.f, 0.f, 0.f, 0.f};
    for (int w = 0; w < WPB; ++w) {
      o.x += sm.red[w][0];
      o.y += sm.red[w][1];
      o.z += sm.red[w][2];
      o.w += sm.red[w][3];
    }
    partials[(size_t)b * NXB + blockIdx.x] = o;
  }
}

__global__ __launch_bounds__(TPB) void instance_loss_finalize(
    const float4* __restrict__ partials,  // [B, NXB]
    const float* __restrict__ ann,        // [B,32,53]
    float* __restrict__ out)              // [3]: act, obj, reg
{
  const int tid  = threadIdx.x;
  const int lane = tid & 31;
  const int wave = tid >> 5;
  __shared__ float sred[WPB][4];

  float accA = 0.f, accO = 0.f, accR = 0.f;  // meaningful on tid 0
  for (int b = 0; b < BB; ++b) {
    float4 p = partials[(size_t)b * NXB + tid];
    float a = p.x, o = p.y, r = p.z, np = p.w;
    a  = wave_sum(a);
    o  = wave_sum(o);
    r  = wave_sum(r);
    np = wave_sum(np);
    if (lane == 0) {
      sred[wave][0] = a; sred[wave][1] = o; sred[wave][2] = r; sred[wave][3] = np;
    }
    __syncthreads();
    if (tid == 0) {
      float A = 0.f, O = 0.f, R = 0.f, P = 0.f;
      for (int w = 0; w < WPB; ++w) {
        A += sred[w][0]; O += sred[w][1]; R += sred[w][2]; P += sred[w][3];
      }
      bool hv = false;
      for (int m = 0; m < MM; ++m)
        hv = hv || (ann[(size_t)b * (MM * ANNW) + m * ANNW + 4] != -1.f);
      const float h     = hv ? 1.f : 0.f;
      const float denom = fmaxf(P, 1.f);
      accA += h * (A / denom);
      accO += h * (O / denom);
      accR += h * ((P > 0.f) ? (R / fmaxf(P * 4.f, 1.f)) : 0.f);
    }
    __syncthreads();
  }
  if (tid == 0) {
    out[0] = accA * (1.f / BB);
    out[1] = accO * (1.f / BB);
    out[2] = accR * (1.f / BB);
  }
}

extern "C" void kernel_launch(void* const* d_in, const int* in_sizes, int n_in,
                              void* d_out, int out_size, void* d_ws, size_t ws_size,
                              hipStream_t stream) {
  const float* act_cls = (const float*)d_in[0];
  const float* obj_cls = (const float*)d_in[1];
  const float* regr    = (const float*)d_in[2];
  const float* anchors = (const float*)d_in[3];
  const float* ann     = (const float*)d_in[4];
  float* out = (float*)d_out;
  float4* partials = (float4*)d_ws;  // 8 * 512 * 16B = 64 KB

  dim3 grid(NXB, BB);
  instance_loss_main<<<grid, TPB, 0, stream>>>(act_cls, obj_cls, regr, anchors, ann, partials);
  instance_loss_finalize<<<1, TPB, 0, stream>>>(partials, ann, out);
}